// ResidualVQ_88021059764282
// MI455X (gfx1250) — compile-verified
//
#include <hip/hip_runtime.h>

typedef float v2f __attribute__((ext_vector_type(2)));
typedef float v8f __attribute__((ext_vector_type(8)));

#define DDIM 128      // feature dim
#define KCB  2048     // codebook entries per stage
#define QST  8        // residual stages
#define MTILE 128     // tokens per workgroup (8 waves x 16 rows)
#define EBLK 32       // codebook entries staged into LDS per iteration
#define LSTR 132      // dwords per entry in LDS (128 + 4 pad) -> conflict-free b64

// ---------------------------------------------------------------------------
// Pass 1: 0.5 * |e_k|^2 for every codebook entry (8*2048 = 16K values -> d_ws)
// ---------------------------------------------------------------------------
__global__ __launch_bounds__(256) void rvq_norms_kernel(
    const float* __restrict__ cb, float* __restrict__ nrm, int total) {
  int i = blockIdx.x * blockDim.x + threadIdx.x;
  if (i >= total) return;
  const float4* p = (const float4*)(cb + (size_t)i * DDIM);
  float s = 0.f;
#pragma unroll
  for (int j = 0; j < DDIM / 4; ++j) {
    float4 v = p[j];
    s += v.x * v.x + v.y * v.y + v.z * v.z + v.w * v.w;
  }
  nrm[i] = 0.5f * s;
}

// ---------------------------------------------------------------------------
// Pass 2: fused 8-stage residual VQ.
// Residual lives in registers in WMMA A-fragment layout for the whole kernel.
// Per staged block and column tile: ALL 32 B fragments are batch-loaded, a
// sched_barrier(0) pins them above the WMMA chain (distinct DS dest quads,
// streaming partial dscnt waits), then 32 V_WMMA_F32_16X16X4_F32 run.
// ---------------------------------------------------------------------------
__global__ __launch_bounds__(256, 1) void rvq_main_kernel(
    const float* __restrict__ x, const float* __restrict__ cb,
    const float* __restrict__ nrm, float* __restrict__ out) {
  __shared__ float cbuf[2][EBLK * LSTR];  // double-buffered codebook tile
  __shared__ int   widx[8 * 16];          // per-wave winning index per row

  const int tid  = threadIdx.x;
  const int wave = tid >> 5;
  const int lane = tid & 31;
  const int m    = lane & 15;   // row within wave tile / column within K tile
  const int h    = lane >> 4;   // half-wave select (K/M split in fragments)
  const int row0 = blockIdx.x * MTILE + wave * 16;
  const int row  = row0 + m;

  // Load residual r = x[row, :] directly into A-fragment layout:
  // lane h*16+m holds r[m, 4*dc + 2h] and r[m, 4*dc + 2h + 1] for dc = 0..31.
  float a[2 * (DDIM / 4)];
#pragma unroll
  for (int dc = 0; dc < DDIM / 4; ++dc) {
    v2f v = *(const v2f*)(x + (size_t)row * DDIM + 4 * dc + 2 * h);
    a[2 * dc + 0] = v[0];
    a[2 * dc + 1] = v[1];
  }

  for (int q = 0; q < QST; ++q) {
    const float* cbq = cb + (size_t)q * KCB * DDIM;
    const float* nq  = nrm + q * KCB;

    float bestV[8];
    int   bestC[8];
#pragma unroll
    for (int v = 0; v < 8; ++v) { bestV[v] = -3.4e38f; bestC[v] = 0; }

    int pb = 0;
    for (int kb = 0; kb < KCB; kb += EBLK, pb ^= 1) {
      // ---- cooperative stage: 32 entries x 128 floats into LDS (b128 path) ----
      {
        const int e  = tid >> 3;         // entry 0..31 (8 threads per entry)
        const int d0 = (tid & 7) * 16;   // 16 floats per thread
        const float4* src = (const float4*)(cbq + (size_t)(kb + e) * DDIM + d0);
        float4* dst = (float4*)&cbuf[pb][e * LSTR + d0];
#pragma unroll
        for (int j = 0; j < 4; ++j) dst[j] = src[j];
        if (kb + EBLK < KCB)  // global_prefetch_b8 of next tile
          __builtin_prefetch(cbq + (size_t)(kb + EBLK + e) * DDIM + d0, 0, 0);
      }
      __syncthreads();

      const float* lb = &cbuf[pb][0];

      // ---- two 16-entry column tiles; per tile: batch B loads then WMMAs ----
#pragma unroll
      for (int ct = 0; ct < 2; ++ct) {
        const int off = (ct * 16 + m) * LSTR + 2 * h;
        const int col = kb + ct * 16 + m;
        const float nv = nq[col];  // 0.5*|e_col|^2 (L2-resident scalar)

        // Batch-issue ALL 32 B fragments (16x ds_load_2addr_b64).
        v2f bfr[DDIM / 4];
#pragma unroll
        for (int dc = 0; dc < DDIM / 4; ++dc)
          bfr[dc] = *(const v2f*)(lb + off + 4 * dc);

        // Hard scheduling fence: loads may not sink below, WMMAs may not
        // hoist above -> all fragments live here, distinct DS dest regs,
        // in-order DS returns stream into the WMMA chain via partial waits.
        __builtin_amdgcn_sched_barrier(0);

        // 32 back-to-back WMMAs, no per-pair full dscnt stalls.
        v8f c = {};
#pragma unroll
        for (int dc = 0; dc < DDIM / 4; ++dc) {
          v2f av; av[0] = a[2 * dc]; av[1] = a[2 * dc + 1];
          c = __builtin_amdgcn_wmma_f32_16x16x4_f32(
              false, av, false, bfr[dc], (short)0, c, false, false);
        }

        // fused argmax: strict '>' + increasing column order == first-tie rule
#pragma unroll
        for (int v = 0; v < 8; ++v) {
          float s = c[v] - nv;  // argmax(r.e - 0.5|e|^2) == argmin distance
          if (s > bestV[v]) { bestV[v] = s; bestC[v] = col; }
        }
      }
    }

    // ---- argmax reduction across the 16 lanes of each half-wave ----
#pragma unroll
    for (int off = 1; off < 16; off <<= 1) {
#pragma unroll
      for (int v = 0; v < 8; ++v) {
        float ov = __shfl_xor(bestV[v], off, 32);
        int   oc = __shfl_xor(bestC[v], off, 32);
        if (ov > bestV[v] || (ov == bestV[v] && oc < bestC[v])) {
          bestV[v] = ov; bestC[v] = oc;
        }
      }
    }
    // lane 0 holds rows 0..7, lane 16 holds rows 8..15 of this wave's tile
    if (m == 0) {
#pragma unroll
      for (int v = 0; v < 8; ++v) widx[wave * 16 + 8 * h + v] = bestC[v];
    }
    __syncthreads();

    const int idx = widx[wave * 16 + m];

    // emit code (output dtype convention: float)
    if (h == 0) out[(size_t)row * QST + q] = (float)idx;

    // gather winning codeword and subtract from register-resident residual
    const float* ev = cbq + (size_t)idx * DDIM;
#pragma unroll
    for (int dc = 0; dc < DDIM / 4; ++dc) {
      v2f g = *(const v2f*)(ev + 4 * dc + 2 * h);
      a[2 * dc + 0] -= g[0];
      a[2 * dc + 1] -= g[1];
    }
  }
}

extern "C" void kernel_launch(void* const* d_in, const int* in_sizes, int n_in,
                              void* d_out, int out_size, void* d_ws, size_t ws_size,
                              hipStream_t stream) {
  const float* x   = (const float*)d_in[0];   // [B,T,D] f32
  const float* cb  = (const float*)d_in[1];   // [Q,K,D] f32
  float*       out = (float*)d_out;           // [B,T,Q] codes (as float)
  float*       nrm = (float*)d_ws;            // Q*K = 16384 floats (64 KB)

  const int N = in_sizes[0] / DDIM;           // 65536 tokens
  const int totalEntries = QST * KCB;

  rvq_norms_kernel<<<(totalEntries + 255) / 256, 256, 0, stream>>>(cb, nrm, totalEntries);
  rvq_main_kernel<<<N / MTILE, 256, 0, stream>>>(x, cb, nrm, out);
}